// QuantumStatePreparation_51668456571056
// MI455X (gfx1250) — compile-verified
//
#include <hip/hip_runtime.h>
#include <math.h>

typedef float v2f __attribute__((ext_vector_type(2)));
typedef float v8f __attribute__((ext_vector_type(8)));

#define BATCH   4096
#define FEAT    256
#define HID     48
#define HILBERT 4096
#define TWOH    8192
#define OUTW    4098
#define EPSF    1e-12f
// -(HILBERT-1)*EPS*log(EPS)
#define ZERO_EIG_TERM 1.1314903e-07f
#define NSWEEPS 10

__device__ __forceinline__ void cmul(float ar, float ai, float br, float bi,
                                     float& cr, float& ci) {
  cr = ar * br - ai * bi;
  ci = ar * bi + ai * br;
}

// ---------------------------------------------------------------------------
// Kernel 0: compose the 33 2x2 complex gates into one matrix T (row-vector
// convention: s' = s @ T, T = R_0 R_1 ... R_32).
// ---------------------------------------------------------------------------
__global__ void k_compose_gate(const float* __restrict__ ent, float* __restrict__ R) {
  if (threadIdx.x != 0 || blockIdx.x != 0) return;
  float t00r = 1.f, t00i = 0.f, t01r = 0.f, t01i = 0.f;
  float t10r = 0.f, t10i = 0.f, t11r = 1.f, t11i = 0.f;
  for (int d = 0; d < 3; ++d) {
    for (int q = 0; q < 11; ++q) {  // entanglement_params[:, :NUM_QUBITS-1, :]
      const float* p = ent + (d * 12 + q) * 4;
      float th = p[0], ph = p[1], la = p[2], ga = p[3];
      float c = cosf(0.5f * th), s = sinf(0.5f * th);
      float g00r = c,            g00i = 0.f;
      float g01r = -cosf(la)*s,  g01i = -sinf(la)*s;
      float g10r =  cosf(ph)*s,  g10i =  sinf(ph)*s;
      float g11r =  cosf(ga)*c,  g11i =  sinf(ga)*c;
      float xr, xi, yr, yi;
      float n00r, n00i, n01r, n01i, n10r, n10i, n11r, n11i;
      cmul(t00r,t00i,g00r,g00i,xr,xi); cmul(t01r,t01i,g10r,g10i,yr,yi); n00r=xr+yr; n00i=xi+yi;
      cmul(t00r,t00i,g01r,g01i,xr,xi); cmul(t01r,t01i,g11r,g11i,yr,yi); n01r=xr+yr; n01i=xi+yi;
      cmul(t10r,t10i,g00r,g00i,xr,xi); cmul(t11r,t11i,g10r,g10i,yr,yi); n10r=xr+yr; n10i=xi+yi;
      cmul(t10r,t10i,g01r,g01i,xr,xi); cmul(t11r,t11i,g11r,g11i,yr,yi); n11r=xr+yr; n11i=xi+yi;
      t00r=n00r; t00i=n00i; t01r=n01r; t01i=n01i;
      t10r=n10r; t10i=n10i; t11r=n11r; t11i=n11i;
    }
  }
  R[0]=t00r; R[1]=t00i; R[2]=t01r; R[3]=t01i;
  R[4]=t10r; R[5]=t10i; R[6]=t11r; R[7]=t11i;
}

// ---------------------------------------------------------------------------
// Kernel 1: h = tanh(X @ W1 + b1), [4096,256]x[256,48] via wmma_f32_16x16x4.
// One wave per 16 batch rows; 3 N-tiles; K=256 -> 64 k-steps.
// ---------------------------------------------------------------------------
__global__ __launch_bounds__(256) void k_gemm1(const float* __restrict__ X,
                                               const float* __restrict__ W1,
                                               const float* __restrict__ b1,
                                               float* __restrict__ hout) {
  int tid  = threadIdx.x;
  int wave = tid >> 5;
  int lane = tid & 31;
  int lh   = lane >> 4;          // lane-half selects K+2 (A) / K+2 (B)
  int l15  = lane & 15;
  int koff = lh * 2;
  int rowBase = (blockIdx.x * 8 + wave) * 16;
  int row = rowBase + l15;

  v8f cacc[3];
  for (int nt = 0; nt < 3; ++nt)
    for (int v = 0; v < 8; ++v) cacc[nt][v] = 0.f;

  for (int kk = 0; kk < 64; ++kk) {
    v2f a;
    a.x = X[row * FEAT + 4 * kk + koff];
    a.y = X[row * FEAT + 4 * kk + koff + 1];
    for (int nt = 0; nt < 3; ++nt) {
      int col = nt * 16 + l15;
      v2f b;
      b.x = W1[(4 * kk + koff) * HID + col];
      b.y = W1[(4 * kk + koff + 1) * HID + col];
      cacc[nt] = __builtin_amdgcn_wmma_f32_16x16x4_f32(
          false, a, false, b, (short)0, cacc[nt], false, false);
    }
  }
  for (int nt = 0; nt < 3; ++nt) {
    int col = nt * 16 + l15;
    float bias = b1[col];
    for (int v = 0; v < 8; ++v) {
      int r = rowBase + v + 8 * lh;
      hout[r * HID + col] = tanhf(cacc[nt][v] + bias);
    }
  }
}

// ---------------------------------------------------------------------------
// Kernel 2: amps = h @ W2 + b2, [4096,48]x[48,8192] via wmma (K=48 -> 12
// k-steps). Block = 16 batch rows x 8 waves over 512 column tiles. Fused
// per-row sum-of-squares -> norms.
// ---------------------------------------------------------------------------
__global__ __launch_bounds__(256) void k_gemm2(const float* __restrict__ h,
                                               const float* __restrict__ W2,
                                               const float* __restrict__ b2,
                                               float* __restrict__ amps,
                                               float* __restrict__ norms) {
  __shared__ float hs[16 * HID];
  __shared__ float rowsum[16];
  int tid = threadIdx.x;
  int rowBase = blockIdx.x * 16;
  for (int i = tid; i < 16 * HID; i += 256) hs[i] = h[rowBase * HID + i];
  if (tid < 16) rowsum[tid] = 0.f;
  __syncthreads();

  int wave = tid >> 5;
  int lane = tid & 31;
  int lh   = lane >> 4;
  int l15  = lane & 15;
  int koff = lh * 2;

  v2f a[12];
  for (int kk = 0; kk < 12; ++kk) {
    a[kk].x = hs[l15 * HID + 4 * kk + koff];
    a[kk].y = hs[l15 * HID + 4 * kk + koff + 1];
  }

  float acc[8];
  for (int v = 0; v < 8; ++v) acc[v] = 0.f;

  for (int nt = wave; nt < 512; nt += 8) {
    int col = nt * 16 + l15;
    v8f c;
    for (int v = 0; v < 8; ++v) c[v] = 0.f;
    for (int kk = 0; kk < 12; ++kk) {
      v2f b;
      b.x = W2[(4 * kk + koff) * TWOH + col];
      b.y = W2[(4 * kk + koff + 1) * TWOH + col];
      c = __builtin_amdgcn_wmma_f32_16x16x4_f32(
          false, a[kk], false, b, (short)0, c, false, false);
    }
    float bias = b2[col];
    for (int v = 0; v < 8; ++v) {
      float val = c[v] + bias;
      amps[(size_t)(rowBase + v + 8 * lh) * TWOH + col] = val;
      acc[v] += val * val;
    }
  }
  // reduce sum-of-squares across the 16 lanes of each half
  for (int v = 0; v < 8; ++v) {
    float x = acc[v];
    x += __shfl_xor(x, 1, 32);
    x += __shfl_xor(x, 2, 32);
    x += __shfl_xor(x, 4, 32);
    x += __shfl_xor(x, 8, 32);
    if (l15 == 0) atomicAdd(&rowsum[v + 8 * lh], x);
  }
  __syncthreads();
  if (tid < 16) norms[rowBase + tid] = sqrtf(rowsum[tid]);
}

// ---------------------------------------------------------------------------
// Kernel 3: probs (normalized, gated) + per-row entropy.
// ---------------------------------------------------------------------------
__global__ __launch_bounds__(256) void k_probs(const float* __restrict__ amps,
                                               const float* __restrict__ norms,
                                               const float* __restrict__ R,
                                               float* __restrict__ out) {
  size_t idx = (size_t)blockIdx.x * 256 + threadIdx.x;  // 0 .. 16M-1
  int b = (int)(idx >> 12);
  int j = (int)(idx & 4095);
  const float* row = amps + (size_t)b * TWOH;
  float nm = norms[b];
  float scale = 1.f / fmaxf(nm, EPSF);
  float re = row[j];
  float im = row[HILBERT + j];
  if (j < 2) {
    float a0r = row[0], a0i = row[HILBERT];
    float a1r = row[1], a1i = row[HILBERT + 1];
    float g0r, g0i, g1r, g1i, xr, xi, yr, yi;
    cmul(a0r, a0i, R[0], R[1], xr, xi); cmul(a1r, a1i, R[4], R[5], yr, yi);
    g0r = xr + yr; g0i = xi + yi;
    cmul(a0r, a0i, R[2], R[3], xr, xi); cmul(a1r, a1i, R[6], R[7], yr, yi);
    g1r = xr + yr; g1i = xi + yi;
    re = (j == 0) ? g0r : g1r;
    im = (j == 0) ? g0i : g1i;
    if (j == 0) {
      // n2 = sum|psi|^2; gate only perturbs elements 0,1
      float n2 = (nm * nm - (a0r*a0r + a0i*a0i + a1r*a1r + a1i*a1i)
                          + (g0r*g0r + g0i*g0i + g1r*g1r + g1i*g1i)) * scale * scale;
      float lam = fmaxf(n2, EPSF);
      out[(size_t)b * OUTW + HILBERT] = -lam * logf(lam) + ZERO_EIG_TERM;
    }
  }
  re *= scale;
  im *= scale;
  out[(size_t)b * OUTW + j] = re * re + im * im;
}

// ---------------------------------------------------------------------------
// Kernel 4: Schmidt entropy. One block per batch row. M = psi.reshape(64,64)
// in LDS; one-sided complex Jacobi (tournament ordering: 63 rounds x 32
// disjoint column pairs, 8 lanes per pair) -> singular values = column norms.
// ---------------------------------------------------------------------------
__global__ __launch_bounds__(256) void k_jacobi(const float* __restrict__ amps,
                                                const float* __restrict__ norms,
                                                const float* __restrict__ R,
                                                float* __restrict__ out) {
  __shared__ float Mre[64 * 65];   // padded stride 65 -> conflict-free columns
  __shared__ float Mim[64 * 65];
  __shared__ float cn[64];

  int tid = threadIdx.x;
  int b = blockIdx.x;
  const float* row = amps + (size_t)b * TWOH;
  float scale = 1.f / fmaxf(norms[b], EPSF);

  for (int j = tid; j < HILBERT; j += 256) {
    int rr = j >> 6, cc = j & 63;
    Mre[rr * 65 + cc] = row[j] * scale;
    Mim[rr * 65 + cc] = row[HILBERT + j] * scale;
  }
  __syncthreads();
  if (tid == 0) {
    float a0r = Mre[0], a0i = Mim[0], a1r = Mre[1], a1i = Mim[1];
    float xr, xi, yr, yi;
    cmul(a0r, a0i, R[0], R[1], xr, xi); cmul(a1r, a1i, R[4], R[5], yr, yi);
    Mre[0] = xr + yr; Mim[0] = xi + yi;
    cmul(a0r, a0i, R[2], R[3], xr, xi); cmul(a1r, a1i, R[6], R[7], yr, yi);
    Mre[1] = xr + yr; Mim[1] = xi + yi;
  }
  __syncthreads();

  // initial column norms: 4 lanes per column, 16 rows each
  {
    int col = tid >> 2, part = tid & 3;
    float s = 0.f;
    for (int k = 0; k < 16; ++k) {
      int rr = part * 16 + k;
      float xr = Mre[rr * 65 + col], xi = Mim[rr * 65 + col];
      s += xr * xr + xi * xi;
    }
    s += __shfl_xor(s, 1, 32);
    s += __shfl_xor(s, 2, 32);
    if (part == 0) cn[col] = s;
  }
  __syncthreads();

  int p  = tid >> 3;   // pair id 0..31
  int r8 = tid & 7;    // lane within pair

  for (int sweep = 0; sweep < NSWEEPS; ++sweep) {
    for (int rnd = 0; rnd < 63; ++rnd) {
      int ci, cj;
      if (p == 0) { ci = 63; cj = rnd; }
      else        { ci = (rnd + p) % 63; cj = (rnd + 63 - p) % 63; }

      float br = 0.f, bi = 0.f;
      for (int k = 0; k < 8; ++k) {
        int rr = r8 + 8 * k;
        float xr = Mre[rr * 65 + ci], xi = Mim[rr * 65 + ci];
        float yr = Mre[rr * 65 + cj], yi = Mim[rr * 65 + cj];
        br += xr * yr + xi * yi;       // Re(conj(ci) . cj)
        bi += xr * yi - xi * yr;       // Im(conj(ci) . cj)
      }
      br += __shfl_xor(br, 1, 32); br += __shfl_xor(br, 2, 32); br += __shfl_xor(br, 4, 32);
      bi += __shfl_xor(bi, 1, 32); bi += __shfl_xor(bi, 2, 32); bi += __shfl_xor(bi, 4, 32);

      float aN = cn[ci], cN = cn[cj];
      float hb = sqrtf(br * br + bi * bi);
      if (hb > 1e-30f) {
        float tau = (cN - aN) / (2.f * hb);
        float tt  = (tau >= 0.f ? 1.f : -1.f) / (fabsf(tau) + sqrtf(1.f + tau * tau));
        float cs  = rsqrtf(1.f + tt * tt);
        float sn  = tt * cs;
        float wr  = br / hb, wi = -bi / hb;  // e^{-i beta}
        for (int k = 0; k < 8; ++k) {
          int rr = r8 + 8 * k;
          float xr = Mre[rr * 65 + ci], xi = Mim[rr * 65 + ci];
          float yr = Mre[rr * 65 + cj], yi = Mim[rr * 65 + cj];
          float tr = yr * wr - yi * wi;
          float ti = yr * wi + yi * wr;
          Mre[rr * 65 + ci] = cs * xr - sn * tr;
          Mim[rr * 65 + ci] = cs * xi - sn * ti;
          Mre[rr * 65 + cj] = sn * xr + cs * tr;
          Mim[rr * 65 + cj] = sn * xi + cs * ti;
        }
        if (r8 == 0) {
          float d = 2.f * cs * sn * hb;
          cn[ci] = cs * cs * aN + sn * sn * cN - d;
          cn[cj] = sn * sn * aN + cs * cs * cN + d;
        }
      }
      __syncthreads();
    }
  }

  // exact final column norms (sigma^2)
  {
    int col = tid >> 2, part = tid & 3;
    float s = 0.f;
    for (int k = 0; k < 16; ++k) {
      int rr = part * 16 + k;
      float xr = Mre[rr * 65 + col], xi = Mim[rr * 65 + col];
      s += xr * xr + xi * xi;
    }
    s += __shfl_xor(s, 1, 32);
    s += __shfl_xor(s, 2, 32);
    if (part == 0) cn[col] = s;
  }
  __syncthreads();
  if (tid == 0) {
    float s = 0.f;
    for (int jj = 0; jj < 64; ++jj) {
      float l = fmaxf(cn[jj], 1e-24f);  // max(S,EPS)^2 == max(S^2, EPS^2)
      s += l * logf(l);
    }
    out[(size_t)b * OUTW + HILBERT + 1] = -s;
  }
}

// ---------------------------------------------------------------------------
extern "C" void kernel_launch(void* const* d_in, const int* in_sizes, int n_in,
                              void* d_out, int out_size, void* d_ws, size_t ws_size,
                              hipStream_t stream) {
  (void)in_sizes; (void)n_in; (void)out_size; (void)ws_size;
  const float* X   = (const float*)d_in[0];
  const float* W1  = (const float*)d_in[1];
  const float* b1  = (const float*)d_in[2];
  const float* W2  = (const float*)d_in[3];
  const float* b2  = (const float*)d_in[4];
  const float* ent = (const float*)d_in[5];
  float* out = (float*)d_out;

  float* ws    = (float*)d_ws;
  float* R     = ws;                         // 8 floats
  float* norms = ws + 8;                     // 4096 floats
  float* h     = ws + 8 + BATCH;             // 4096*48 floats
  float* amps  = h + (size_t)BATCH * HID;    // 4096*8192 floats (~128 MB)

  hipLaunchKernelGGL(k_compose_gate, dim3(1), dim3(32), 0, stream, ent, R);
  hipLaunchKernelGGL(k_gemm1, dim3(BATCH / 128), dim3(256), 0, stream, X, W1, b1, h);
  hipLaunchKernelGGL(k_gemm2, dim3(BATCH / 16), dim3(256), 0, stream, h, W2, b2, amps, norms);
  hipLaunchKernelGGL(k_probs, dim3((unsigned)(((size_t)BATCH * HILBERT) / 256)), dim3(256), 0, stream,
                     amps, norms, R, out);
  hipLaunchKernelGGL(k_jacobi, dim3(BATCH), dim3(256), 0, stream, amps, norms, R, out);
}